// OHEMLoss_35012573397261
// MI455X (gfx1250) — compile-verified
//
#include <hip/hip_runtime.h>
#include <hip/hip_bf16.h>

// ---------------- problem constants (fixed by the reference) ----------------
#define IGNORE_INDEX 255
#define MIN_KEPT     100000
#define KEEP_RATIO_F 0.3f

#define C_CLS 19
#define TPB   256
#define HW_TOT (512 * 1024)          // H*W per image
#define RBLK  1024                   // blocks for streaming passes

#define AS1 __attribute__((address_space(1)))
#define AS3 __attribute__((address_space(3)))

typedef __attribute__((__vector_size__(4 * sizeof(int)))) int v4i_t;

// CDNA5 async global->LDS path (gfx1250) — probe-verified to lower to
// global_load_async_to_lds_b128 + s_wait_asynccnt on this toolchain.
#if defined(__has_builtin)
#  if __has_builtin(__builtin_amdgcn_global_load_async_to_lds_b128) && \
      __has_builtin(__builtin_amdgcn_s_wait_asynccnt)
#    define USE_ASYNC_LDS 1
#  endif
#endif
#ifndef USE_ASYNC_LDS
#  define USE_ASYNC_LDS 0
#endif

// order-preserving float -> uint mapping (monotonic increasing)
__device__ __forceinline__ unsigned f2ord(float f) {
    unsigned s = __float_as_uint(f);
    return (s & 0x80000000u) ? ~s : (s | 0x80000000u);
}
__device__ __forceinline__ float ord2f(unsigned u) {
    unsigned s = (u & 0x80000000u) ? (u & 0x7fffffffu) : ~u;
    return __uint_as_float(s);
}

// ctrl words (uint32):
// [0]=n_valid [1]=n_keep [2]=flag(threshold search needed)
// [3]=prefix (selected high digits of ordered bits) [4]=remaining rank [6]=thr bits

// ---------------------------------------------------------------------------
// Pass 1: per-pixel CE loss via async global->LDS staging. No atomics.
// ---------------------------------------------------------------------------
__global__ void k_loss(const float* __restrict__ logits,
                       const int* __restrict__ labels,
                       float* __restrict__ loss,
                       int* __restrict__ vcnt)
{
    __shared__ __align__(16) float tile[C_CLS * TPB];   // [c][pixel], 19 KB
    __shared__ int wsum[TPB / 32];

    const int tid = threadIdx.x;
    const int blk = blockIdx.x;
    const int blocksPerImg = HW_TOT / TPB;              // 2048
    const int b   = blk / blocksPerImg;
    const int hwb = (blk % blocksPerImg) * TPB;

    const size_t pix = (size_t)b * HW_TOT + hwb + tid;  // flat pixel index
    __builtin_prefetch(&labels[pix], 0, 1);             // global_prefetch_b8

    const float* src_base = logits + ((size_t)b * C_CLS) * (size_t)HW_TOT + hwb;

#if USE_ASYNC_LDS
    {
        // 19 rows x 1024B; each row = 64 lanes x 16B (b128); 4 rows per round.
        const int j  = tid & 63;     // 16B chunk within row
        const int r0 = tid >> 6;     // row within round
        #pragma unroll
        for (int rnd = 0; rnd < 5; ++rnd) {
            const int c = rnd * 4 + r0;
            if (c < C_CLS) {
                const float* g = src_base + (size_t)c * HW_TOT + (size_t)j * 4;
                AS1 v4i_t* gp = (AS1 v4i_t*)(unsigned long long)(size_t)g;
                AS3 v4i_t* lp = (AS3 v4i_t*)&tile[c * TPB + j * 4];
                __builtin_amdgcn_global_load_async_to_lds_b128(gp, lp, 0, 0);
            }
        }
        __builtin_amdgcn_s_wait_asynccnt(0);
    }
    __syncthreads();
#else
    for (int idx = tid; idx < C_CLS * (TPB / 4); idx += TPB) {
        const int c = idx >> 6;
        const int j = idx & 63;
        ((float4*)&tile[c * TPB])[j] =
            ((const float4*)(src_base + (size_t)c * HW_TOT))[j];
    }
    __syncthreads();
#endif

    const int  lab   = labels[pix];
    const bool valid = (lab != IGNORE_INDEX);
    int slab = valid ? lab : 0;
    if (slab < 0 || slab >= C_CLS) slab = 0;

    float maxv = -INFINITY;
    #pragma unroll
    for (int c = 0; c < C_CLS; ++c) maxv = fmaxf(maxv, tile[c * TPB + tid]);
    float sum = 0.0f;
    #pragma unroll
    for (int c = 0; c < C_CLS; ++c) sum += __expf(tile[c * TPB + tid] - maxv);
    const float l = __logf(sum) + maxv - tile[slab * TPB + tid];

    // NaN marks ignored pixels (excluded everywhere downstream)
    loss[pix] = valid ? l : __uint_as_float(0x7fc00000u);

    // contention-free valid count: ballot -> per-wave popc -> per-block sum
    const unsigned long long m = __ballot(valid);
    if ((tid & 31) == 0) wsum[tid >> 5] = __popcll(m);
    __syncthreads();
    if (tid == 0) {
        int s = 0;
        #pragma unroll
        for (int w = 0; w < TPB / 32; ++w) s += wsum[w];
        vcnt[blk] = s;
    }
}

// ---------------------------------------------------------------------------
// n_valid / n_keep setup (1 block)
// ---------------------------------------------------------------------------
__global__ void k_prep(const int* __restrict__ vcnt, int nblk, unsigned* ctrl)
{
    __shared__ int part[TPB];
    const int t = threadIdx.x;
    int s = 0;
    for (int i = t; i < nblk; i += TPB) s += vcnt[i];
    part[t] = s;
    __syncthreads();
    for (int o = TPB / 2; o > 0; o >>= 1) {
        if (t < o) part[t] += part[t + o];
        __syncthreads();
    }
    if (t == 0) {
        const int nv = part[0];
        int nk = (int)(KEEP_RATIO_F * (float)nv);       // f32 mult, truncate (matches ref)
        const int m = nv < MIN_KEPT ? nv : MIN_KEPT;
        if (nk < m)  nk = m;
        if (nk > nv) nk = nv;
        const int flag = (nk > 0 && nk < nv) ? 1 : 0;
        ctrl[0] = (unsigned)nv;
        ctrl[1] = (unsigned)nk;
        ctrl[2] = (unsigned)flag;
        ctrl[3] = 0u;                                   // prefix
        ctrl[4] = (unsigned)nk;                         // remaining rank
        if (!flag) ctrl[6] = __float_as_uint(-INFINITY);
    }
}

// ---------------------------------------------------------------------------
// 8-bit-digit radix-select histogram pass, LDS-privatized (skew-proof)
// ---------------------------------------------------------------------------
__global__ void k_hist8(const float4* __restrict__ loss4, int n4,
                        unsigned* __restrict__ hist,
                        const unsigned* __restrict__ ctrl, int lvl)
{
    if (ctrl[2] == 0u) return;                           // uniform
    __shared__ unsigned lh[256];
    lh[threadIdx.x] = 0u;
    __syncthreads();

    const unsigned prefix = ctrl[3];
    const int pshift = 32 - 8 * lvl;                     // prefix bits (lvl>=1)
    const int dshift = 24 - 8 * lvl;                     // digit position
    const int stride = gridDim.x * blockDim.x;
    for (int i = blockIdx.x * blockDim.x + threadIdx.x; i < n4; i += stride) {
        const float4 v = loss4[i];
        const float e0 = v.x, e1 = v.y, e2 = v.z, e3 = v.w;
        #pragma unroll
        for (int e = 0; e < 4; ++e) {
            const float l = (e == 0) ? e0 : (e == 1) ? e1 : (e == 2) ? e2 : e3;
            if (l == l) {                                // !NaN -> valid
                const unsigned u = f2ord(l);
                if (lvl == 0 || (u >> pshift) == prefix)
                    atomicAdd(&lh[(u >> dshift) & 0xffu], 1u);
            }
        }
    }
    __syncthreads();
    const unsigned h = lh[threadIdx.x];
    if (h) atomicAdd(&hist[threadIdx.x], h);             // 256 spread addresses
}

// pick the digit containing the k-th largest; update prefix/rank (1 block)
__global__ void k_sel8(const unsigned* __restrict__ hist, unsigned* ctrl, int lvl)
{
    if (ctrl[2] == 0u) return;                           // uniform
    __shared__ unsigned h[256];
    __shared__ unsigned suf[256];
    const int t = threadIdx.x;
    h[t] = hist[t];
    __syncthreads();
    if (t == 0) {
        unsigned acc = 0;
        for (int i = 255; i >= 0; --i) { suf[i] = acc; acc += h[i]; }
    }
    __syncthreads();
    const unsigned k = ctrl[4];
    const unsigned above = suf[t];
    if (above < k && above + h[t] >= k) {                // exactly one thread
        const unsigned newpref = (ctrl[3] << 8) | (unsigned)t;
        ctrl[3] = newpref;
        ctrl[4] = k - above;
        if (lvl == 3) ctrl[6] = __float_as_uint(ord2f(newpref));
    }
}

// ---------------------------------------------------------------------------
// deterministic sum/count of kept losses (no float atomics)
// ---------------------------------------------------------------------------
__global__ void k_reduce(const float4* __restrict__ loss4,
                         const unsigned* __restrict__ ctrl, int n4,
                         float* __restrict__ psum, int* __restrict__ pcnt)
{
    __shared__ float ssum[TPB];
    __shared__ int   scnt[TPB];
    const float thr = __uint_as_float(ctrl[6]);
    const int t = threadIdx.x;
    float s = 0.0f; int c = 0;
    const int stride = gridDim.x * blockDim.x;
    for (int i = blockIdx.x * blockDim.x + t; i < n4; i += stride) {
        const float4 v = loss4[i];
        const float e0 = v.x, e1 = v.y, e2 = v.z, e3 = v.w;
        if (e0 == e0 && e0 >= thr) { s += e0; ++c; }
        if (e1 == e1 && e1 >= thr) { s += e1; ++c; }
        if (e2 == e2 && e2 >= thr) { s += e2; ++c; }
        if (e3 == e3 && e3 >= thr) { s += e3; ++c; }
    }
    ssum[t] = s; scnt[t] = c;
    __syncthreads();
    for (int o = TPB / 2; o > 0; o >>= 1) {
        if (t < o) { ssum[t] += ssum[t + o]; scnt[t] += scnt[t + o]; }
        __syncthreads();
    }
    if (t == 0) { psum[blockIdx.x] = ssum[0]; pcnt[blockIdx.x] = scnt[0]; }
}

__global__ void k_final(const float* __restrict__ psum,
                        const int* __restrict__ pcnt,
                        int nblk, float* __restrict__ out)
{
    __shared__ float ssum[TPB];
    __shared__ int   scnt[TPB];
    const int t = threadIdx.x;
    float s = 0.0f; int c = 0;
    for (int i = t; i < nblk; i += TPB) { s += psum[i]; c += pcnt[i]; }
    ssum[t] = s; scnt[t] = c;
    __syncthreads();
    for (int o = TPB / 2; o > 0; o >>= 1) {
        if (t < o) { ssum[t] += ssum[t + o]; scnt[t] += scnt[t + o]; }
        __syncthreads();
    }
    if (t == 0) out[0] = ssum[0] / fmaxf((float)scnt[0], 1.0f);
}

// ---------------------------------------------------------------------------
extern "C" void kernel_launch(void* const* d_in, const int* in_sizes, int n_in,
                              void* d_out, int out_size, void* d_ws, size_t ws_size,
                              hipStream_t stream)
{
    const float* logits = (const float*)d_in[0];
    const int*   labels = (const int*)d_in[1];
    const int N  = in_sizes[1];                // pixels = 8*512*1024
    const int n4 = N / 4;
    const int nblocks = N / TPB;               // 16384, exact for these shapes

    // workspace layout
    char*     ws    = (char*)d_ws;
    float*    loss  = (float*)ws;                              // N floats
    int*      vcnt  = (int*)(ws + (size_t)N * 4);              // nblocks ints
    unsigned* hist  = (unsigned*)(vcnt + nblocks);             // 4 x 256 bins
    unsigned* ctrl  = hist + 4 * 256;                          // 16 words
    float*    psum  = (float*)(ctrl + 16);                     // RBLK floats
    int*      pcnt  = (int*)(psum + RBLK);                     // RBLK ints

    // zero hist levels + ctrl each call (harness does not re-poison replays)
    (void)hipMemsetAsync(hist, 0, (4 * 256 + 16) * sizeof(unsigned), stream);

    k_loss <<<nblocks, TPB, 0, stream>>>(logits, labels, loss, vcnt);
    k_prep <<<1,       TPB, 0, stream>>>(vcnt, nblocks, ctrl);
    for (int lvl = 0; lvl < 4; ++lvl) {
        k_hist8<<<RBLK, TPB, 0, stream>>>((const float4*)loss, n4,
                                          hist + lvl * 256, ctrl, lvl);
        k_sel8 <<<1,    TPB, 0, stream>>>(hist + lvl * 256, ctrl, lvl);
    }
    k_reduce<<<RBLK, TPB, 0, stream>>>((const float4*)loss, ctrl, n4, psum, pcnt);
    k_final <<<1,    TPB, 0, stream>>>(psum, pcnt, RBLK, (float*)d_out);
}